// SplineconvModel_82566451298752
// MI455X (gfx1250) — compile-verified
//
#include <hip/hip_runtime.h>
#include <hip/hip_bf16.h>

#define IN_CH  64
#define OUT_CH 64
#define KKNOTS 2

typedef float v2f __attribute__((ext_vector_type(2)));
typedef float v8f __attribute__((ext_vector_type(8)));

// ---------------------------------------------------------------------------
// Kernel 1: xw[k] = x @ weight[k]  (k=0,1),  xr = x @ root + bias
// One wave (32 threads) per 16-row tile of x. V_WMMA_F32_16X16X4_F32 is exact
// f32 FMA math — identical numerics to the reference einsum.
//
// ISA VGPR layouts (CDNA5 ISA):
//   A 16x4 f32 : lane<16 -> row M=lane,     elems K=k0+0,k0+1
//                lane>=16 -> row M=lane-16, elems K=k0+2,k0+3
//   B 4x16 f32 : vgpr0 = row K=k0+{0|2} (per lane half), col N=lane%16
//                vgpr1 = row K=k0+{1|3},                 col N=lane%16
//   C/D 16x16  : vgpr i -> row M=i (lanes 0-15) / M=8+i (lanes 16-31),
//                col N=lane%16
// ---------------------------------------------------------------------------
__global__ __launch_bounds__(32) void spline_gemm_kernel(
    const float* __restrict__ x,       // N x 64
    const float* __restrict__ weight,  // 2 x 64 x 64
    const float* __restrict__ root,    // 64 x 64
    const float* __restrict__ bias,    // 64
    float* __restrict__ xw,            // 2 x N x 64
    float* __restrict__ xr,            // N x 64
    int N)
{
    const int lane = threadIdx.x & 31;
    const int half = lane >> 4;   // 0: lanes 0-15, 1: lanes 16-31
    const int l16  = lane & 15;
    const int row0 = blockIdx.x * 16;

    // Wave-uniform: whole 16-row tile in range? (true for every block when
    // N % 16 == 0; only the last block can take the masked path otherwise)
    const bool full = (row0 + 16 <= N);

    int arow = row0 + l16;
    if (arow >= N) arow = N - 1;          // branchless clamp for tail loads

    // Preload all 16 A fragments (K-steps of 4) for this 16-row tile.
    v2f afrag[16];
    const float* xp = x + (size_t)arow * IN_CH + half * 2;
#pragma unroll
    for (int kk = 0; kk < 16; ++kk) {
        afrag[kk].x = xp[kk * 4 + 0];
        afrag[kk].y = xp[kk * 4 + 1];
    }

#pragma unroll
    for (int m = 0; m < 3; ++m) {
        const float* B  = (m == 0) ? weight
                        : (m == 1) ? (weight + IN_CH * OUT_CH)
                                   : root;
        float* Out      = (m == 0) ? xw
                        : (m == 1) ? (xw + (size_t)N * OUT_CH)
                                   : xr;
#pragma unroll
        for (int nt = 0; nt < 4; ++nt) {
            const int n0 = nt * 16;
            v8f acc = {0.f, 0.f, 0.f, 0.f, 0.f, 0.f, 0.f, 0.f};
#pragma unroll
            for (int kk = 0; kk < 16; ++kk) {
                const float* bp = B + (size_t)(kk * 4 + half * 2) * OUT_CH + n0 + l16;
                v2f bfrag;
                bfrag.x = bp[0];
                bfrag.y = bp[OUT_CH];
                acc = __builtin_amdgcn_wmma_f32_16x16x4_f32(
                    /*neg_a=*/false, afrag[kk],
                    /*neg_b=*/false, bfrag,
                    /*c_mod=*/(short)0, acc,
                    /*reuse_a=*/false, /*reuse_b=*/false);
            }
            // Bias only on the x@root pass; m is compile-time so no +0 adds.
            float* outp = Out + (size_t)(row0 + half * 8) * OUT_CH + n0 + l16;
            if (m == 2) {
                const float bv = bias[n0 + l16];
                if (full) {
#pragma unroll
                    for (int i = 0; i < 8; ++i)
                        outp[(size_t)i * OUT_CH] = acc[i] + bv;
                } else {
#pragma unroll
                    for (int i = 0; i < 8; ++i)
                        if (row0 + half * 8 + i < N)
                            outp[(size_t)i * OUT_CH] = acc[i] + bv;
                }
            } else {
                if (full) {
#pragma unroll
                    for (int i = 0; i < 8; ++i)
                        outp[(size_t)i * OUT_CH] = acc[i];
                } else {
#pragma unroll
                    for (int i = 0; i < 8; ++i)
                        if (row0 + half * 8 + i < N)
                            outp[(size_t)i * OUT_CH] = acc[i];
                }
            }
        }
    }
}

// ---------------------------------------------------------------------------
// Kernel 2: per-edge gather/blend/scatter.  16 lanes per edge, float4 each.
// ---------------------------------------------------------------------------
__global__ __launch_bounds__(256) void spline_edge_kernel(
    const long long* __restrict__ edge_index,  // 2 x E (int64)
    const float* __restrict__ edge_attr,       // E
    const float* __restrict__ xw,              // 2 x N x 64
    float* __restrict__ agg,                   // N x 64 (zeroed)
    float* __restrict__ cnt,                   // N      (zeroed)
    int E, int N)
{
    const int gid = blockIdx.x * blockDim.x + threadIdx.x;
    const int e = gid >> 4;
    const int q = gid & 15;
    if (e >= E) return;

    const long long src = edge_index[e];
    const long long dst = edge_index[(size_t)E + e];

    const float v  = edge_attr[e] * (float)(KKNOTS - 1);
    const float fl = floorf(v);
    int i0 = (int)fl;
    i0 = i0 < 0 ? 0 : (i0 > KKNOTS - 1 ? KKNOTS - 1 : i0);
    const int i1 = (i0 + 1 > KKNOTS - 1) ? KKNOTS - 1 : i0 + 1;
    const float f  = v - fl;
    const float w0 = 1.0f - f;

    const float4 a = ((const float4*)(xw + ((size_t)i0 * N + (size_t)src) * OUT_CH))[q];
    const float4 b = ((const float4*)(xw + ((size_t)i1 * N + (size_t)src) * OUT_CH))[q];

    float4 mv;
    mv.x = w0 * a.x + f * b.x;
    mv.y = w0 * a.y + f * b.y;
    mv.z = w0 * a.z + f * b.z;
    mv.w = w0 * a.w + f * b.w;

    float* ap = agg + (size_t)dst * OUT_CH + q * 4;
    atomicAdd(ap + 0, mv.x);
    atomicAdd(ap + 1, mv.y);
    atomicAdd(ap + 2, mv.z);
    atomicAdd(ap + 3, mv.w);

    if (q == 0) atomicAdd(cnt + dst, 1.0f);
}

// ---------------------------------------------------------------------------
// Kernel 3: out = relu(agg / max(cnt,1) + (x@root + bias))
// ---------------------------------------------------------------------------
__global__ __launch_bounds__(256) void spline_finalize_kernel(
    const float* __restrict__ agg,
    const float* __restrict__ cnt,
    const float* __restrict__ xr,
    float* __restrict__ out,
    int N)
{
    const int gid = blockIdx.x * blockDim.x + threadIdx.x;  // over N*16 float4s
    if (gid >= N * 16) return;
    const int n = gid >> 4;
    float c = cnt[n];
    c = c < 1.0f ? 1.0f : c;
    const float inv = 1.0f / c;

    const float4 a = ((const float4*)agg)[gid];
    const float4 r = ((const float4*)xr)[gid];
    float4 o;
    o.x = fmaxf(a.x * inv + r.x, 0.0f);
    o.y = fmaxf(a.y * inv + r.y, 0.0f);
    o.z = fmaxf(a.z * inv + r.z, 0.0f);
    o.w = fmaxf(a.w * inv + r.w, 0.0f);
    ((float4*)out)[gid] = o;
}

// ---------------------------------------------------------------------------
extern "C" void kernel_launch(void* const* d_in, const int* in_sizes, int n_in,
                              void* d_out, int out_size, void* d_ws, size_t ws_size,
                              hipStream_t stream)
{
    const float*     x      = (const float*)d_in[0];
    const long long* eidx   = (const long long*)d_in[1];  // int64 (2,E)
    const float*     eattr  = (const float*)d_in[2];      // (E,1)
    const float*     weight = (const float*)d_in[3];      // (2,64,64)
    const float*     root   = (const float*)d_in[4];      // (64,64)
    const float*     bias   = (const float*)d_in[5];      // (64,)
    float*           out    = (float*)d_out;

    const int N = in_sizes[0] / IN_CH;   // 50000
    const int E = in_sizes[2];           // 800000

    // Workspace layout: xw (2*N*64) | xr (N*64) | agg (N*64) | cnt (N)
    float* xw  = (float*)d_ws;
    float* xr  = xw  + (size_t)2 * N * OUT_CH;
    float* agg = xr  + (size_t)N * OUT_CH;
    float* cnt = agg + (size_t)N * OUT_CH;

    // Zero the accumulation buffers every call (we accumulate into them).
    hipMemsetAsync(agg, 0, ((size_t)N * OUT_CH + (size_t)N) * sizeof(float), stream);

    const int gemm_blocks = (N + 15) / 16;
    spline_gemm_kernel<<<gemm_blocks, 32, 0, stream>>>(x, weight, root, bias, xw, xr, N);

    const long long edge_threads = (long long)E * 16;
    const int edge_blocks = (int)((edge_threads + 255) / 256);
    spline_edge_kernel<<<edge_blocks, 256, 0, stream>>>(eidx, eattr, xw, agg, cnt, E, N);

    const int fin_threads = N * 16;
    const int fin_blocks = (fin_threads + 255) / 256;
    spline_finalize_kernel<<<fin_blocks, 256, 0, stream>>>(agg, cnt, xr, out, N);
}